// SelfAttention_4114578669640
// MI455X (gfx1250) — compile-verified
//
#include <hip/hip_runtime.h>

// Self-attention for MI455X (gfx1250): N=4, S=2048, E=1024, H=16, D=64.
// All matmuls use V_WMMA_F32_16X16X4_F32 (full fp32, matches fp32 reference).
// wave32 layouts per CDNA5 ISA:
//   A 16x4 : lanes0-15 row L: v0=K0,v1=K1; lanes16-31: v0=K2,v1=K3
//   B 4x16 : v0 = K0(lanes0-15)/K2(lanes16-31), v1 = K1/K3, col = lane%16
//   C 16x16: vgpr i: M=i (lanes0-15) / M=i+8 (lanes16-31), N = lane%16

typedef __attribute__((ext_vector_type(2))) float v2f;
typedef __attribute__((ext_vector_type(4))) float v4f;
typedef __attribute__((ext_vector_type(8))) float v8f;
typedef __attribute__((ext_vector_type(4))) int   v4i;

#define E_DIM 1024
#define S_DIM 2048
#define N_B   4
#define H_N   16
#define D_H   64

static __device__ __forceinline__ v8f wmma_f32(v2f a, v2f b, v8f c) {
  return __builtin_amdgcn_wmma_f32_16x16x4_f32(false, a, false, b, (short)0, c,
                                               false, false);
}

// ---- gfx1250 async global->LDS copy (ASYNCcnt path), with safe fallback ----
#if __has_builtin(__builtin_amdgcn_global_load_async_to_lds_b128)
#define HAVE_ASYNC_LDS 1
#endif

static __device__ __forceinline__ void copy16_to_lds(const float* g, float* l) {
#ifdef HAVE_ASYNC_LDS
  // Param 0 (from hipcc diagnostic): v4i in addrspace(1) — global source.
  // Param 1: v4i in addrspace(3) — LDS destination (byte offset in low 32b).
  typedef __attribute__((address_space(1))) v4i* gvp;
  typedef __attribute__((address_space(3))) v4i* lvp;
  __builtin_amdgcn_global_load_async_to_lds_b128(
      (gvp)(uintptr_t)g, (lvp)(uint32_t)(uintptr_t)l, 0, 0);
#else
  *(v4f*)l = *(const v4f*)g;
#endif
}

static __device__ __forceinline__ void wait_async0() {
  asm volatile("s_wait_asynccnt 0x0" ::: "memory");
}

// ---------------------------------------------------------------------------
// Y[m, c] = sum_k X[m,k] * W[c,k] (+ bias[c])          (torch Linear: x@W.T)
// M = 8192, Nc = 1024, K = 1024. 256 threads (8 waves) per 128x128 tile;
// each wave computes 32x64 (2 A-frags x 4 B-frags -> 8 WMMA per 6 loads).
// ---------------------------------------------------------------------------
__global__ __launch_bounds__(256) void gemm_xwt_kernel(
    const float* __restrict__ X, const float* __restrict__ W,
    float* __restrict__ Y, const float* __restrict__ bias) {
  const int tid  = threadIdx.x;
  const int wave = tid >> 5;
  const int lane = tid & 31;
  const int half = lane >> 4;
  const int lrow = lane & 15;
  const int wm = wave >> 1;          // 0..3
  const int wn = wave & 1;           // 0..1

  const int row0 = blockIdx.x * 128 + wm * 32;
  const int col0 = blockIdx.y * 128 + wn * 64;

  v8f acc[2][4];
#pragma unroll
  for (int mt = 0; mt < 2; ++mt)
#pragma unroll
    for (int ct = 0; ct < 4; ++ct) acc[mt][ct] = (v8f){};

  const float* xp0 = X + (size_t)(row0 + lrow) * E_DIM + 2 * half;
  const float* xp1 = xp0 + (size_t)16 * E_DIM;
  const float* wp0 = W + (size_t)(col0 + lrow) * E_DIM + 2 * half;

#pragma unroll 2
  for (int k = 0; k < E_DIM; k += 4) {
    v2f a0 = *(const v2f*)(xp0 + k);
    v2f a1 = *(const v2f*)(xp1 + k);
    v2f b0 = *(const v2f*)(wp0 + k);
    v2f b1 = *(const v2f*)(wp0 + (size_t)16 * E_DIM + k);
    v2f b2 = *(const v2f*)(wp0 + (size_t)32 * E_DIM + k);
    v2f b3 = *(const v2f*)(wp0 + (size_t)48 * E_DIM + k);
    acc[0][0] = wmma_f32(a0, b0, acc[0][0]);
    acc[0][1] = wmma_f32(a0, b1, acc[0][1]);
    acc[0][2] = wmma_f32(a0, b2, acc[0][2]);
    acc[0][3] = wmma_f32(a0, b3, acc[0][3]);
    acc[1][0] = wmma_f32(a1, b0, acc[1][0]);
    acc[1][1] = wmma_f32(a1, b1, acc[1][1]);
    acc[1][2] = wmma_f32(a1, b2, acc[1][2]);
    acc[1][3] = wmma_f32(a1, b3, acc[1][3]);
  }

#pragma unroll
  for (int ct = 0; ct < 4; ++ct) {
    const int col = col0 + ct * 16 + lrow;
    const float bc = bias ? bias[col] : 0.0f;
#pragma unroll
    for (int i = 0; i < 8; ++i) {
      const int r0 = row0 + i + 8 * half;
      Y[(size_t)r0 * E_DIM + col]        = acc[0][ct][i] + bc;
      Y[(size_t)(r0 + 16) * E_DIM + col] = acc[1][ct][i] + bc;
    }
  }
}

// ---------------------------------------------------------------------------
// Flash attention. 128 threads (4 waves) per (n, h, 64 query rows); each wave
// owns 16 query rows. K/V tiles (16 kv x 64) staged through double-buffered
// LDS via async global->LDS copies shared by all 4 waves.
// ---------------------------------------------------------------------------
#define KV_PAD 68   // 64 floats used, padded: distinct banks + 16B row chunks

__global__ __launch_bounds__(128) void attention_kernel(
    const float* __restrict__ qb, const float* __restrict__ kb,
    const float* __restrict__ vb, const int* __restrict__ mask,
    float* __restrict__ ctx) {
  const int tid  = threadIdx.x;
  const int wave = tid >> 5;
  const int lane = tid & 31;
  const int half = lane >> 4;
  const int lrow = lane & 15;
  const int q0 = blockIdx.x * 64 + wave * 16;
  const int h  = blockIdx.y;
  const int n  = blockIdx.z;

  const float* Q = qb + ((size_t)n * S_DIM + q0) * E_DIM + h * D_H;
  const float* K = kb + (size_t)n * S_DIM * E_DIM + h * D_H;
  const float* V = vb + (size_t)n * S_DIM * E_DIM + h * D_H;

  __shared__ float ldsK[2][16 * KV_PAD];
  __shared__ float ldsV[2][16 * KV_PAD];
  __shared__ float pls[4][16 * 18];

  // Preload Q fragments (16 k-steps over D=64).
  v2f qa[16];
#pragma unroll
  for (int kbs = 0; kbs < 16; ++kbs)
    qa[kbs] = *(const v2f*)(Q + (size_t)lrow * E_DIM + kbs * 4 + 2 * half);

  v8f o[4] = {{}, {}, {}, {}};
  float m_i[8], l_i[8];
#pragma unroll
  for (int i = 0; i < 8; ++i) { m_i[i] = -1e30f; l_i[i] = 0.0f; }

  // Cooperative tile fill: 512 16-byte chunks (K:256, V:256), 4 per thread.
  auto fill_tile = [&](int jj, int bsel) {
#pragma unroll
    for (int it = 0; it < 4; ++it) {
      const int c   = tid + it * 128;
      const int mat = c >> 8;
      const int r   = (c & 255) >> 4;
      const int cc  = c & 15;
      const float* g =
          (mat ? V : K) + (size_t)(jj * 16 + r) * E_DIM + cc * 4;
      float* l = (mat ? &ldsV[bsel][0] : &ldsK[bsel][0]) + r * KV_PAD + cc * 4;
      copy16_to_lds(g, l);
    }
  };

  fill_tile(0, 0);

  const int NT = S_DIM / 16;
  for (int j = 0; j < NT; ++j) {
    const int bsel = j & 1;
    wait_async0();
    __syncthreads();
    if (j + 1 < NT) fill_tile(j + 1, (j + 1) & 1);

    // ---- scores tile S = Q @ K^T (16x16), K-dim = 64, from LDS ----
    v8f s = {};
#pragma unroll
    for (int kbs = 0; kbs < 16; ++kbs) {
      v2f bfr = *(const v2f*)(&ldsK[bsel][lrow * KV_PAD + kbs * 4 + 2 * half]);
      s = wmma_f32(qa[kbs], bfr, s);
    }

    // ---- scale + mask ----
    const int mk = mask[(size_t)n * S_DIM + j * 16 + lrow];
#pragma unroll
    for (int i = 0; i < 8; ++i) {
      float e = s[i] * 0.125f;          // 1/sqrt(64)
      s[i] = (mk == 0) ? -1e20f : e;
    }

    // ---- online softmax row statistics (rows live in 16-lane groups) ----
    float alpha[8];
#pragma unroll
    for (int i = 0; i < 8; ++i) {
      float rm = s[i];
      rm = fmaxf(rm, __shfl_xor(rm, 1));
      rm = fmaxf(rm, __shfl_xor(rm, 2));
      rm = fmaxf(rm, __shfl_xor(rm, 4));
      rm = fmaxf(rm, __shfl_xor(rm, 8));
      const float mn = fmaxf(m_i[i], rm);
      alpha[i] = __expf(m_i[i] - mn);
      m_i[i] = mn;
      const float p = __expf(s[i] - mn);
      s[i] = p;
      float rs = p;
      rs += __shfl_xor(rs, 1);
      rs += __shfl_xor(rs, 2);
      rs += __shfl_xor(rs, 4);
      rs += __shfl_xor(rs, 8);
      l_i[i] = l_i[i] * alpha[i] + rs;
    }
#pragma unroll
    for (int ct = 0; ct < 4; ++ct)
#pragma unroll
      for (int i = 0; i < 8; ++i) o[ct][i] *= alpha[i];

    // ---- stage P (C-layout) via per-wave LDS, reread as A-layout ----
#pragma unroll
    for (int i = 0; i < 8; ++i) pls[wave][(i + 8 * half) * 18 + lrow] = s[i];

#pragma unroll
    for (int kb2 = 0; kb2 < 4; ++kb2) {
      v2f pa = *(const v2f*)(&pls[wave][lrow * 18 + kb2 * 4 + 2 * half]);
      const int vr = (kb2 * 4 + 2 * half) * KV_PAD;
#pragma unroll
      for (int ct = 0; ct < 4; ++ct) {
        v2f bfr;
        bfr.x = ldsV[bsel][vr + ct * 16 + lrow];
        bfr.y = ldsV[bsel][vr + KV_PAD + ct * 16 + lrow];
        o[ct] = wmma_f32(pa, bfr, o[ct]);
      }
    }
  }

  // ---- normalize and store context in [N, S, E] (head-concatenated) ----
  float invl[8];
#pragma unroll
  for (int i = 0; i < 8; ++i) invl[i] = 1.0f / l_i[i];
#pragma unroll
  for (int ct = 0; ct < 4; ++ct)
#pragma unroll
    for (int i = 0; i < 8; ++i)
      ctx[((size_t)n * S_DIM + q0 + i + 8 * half) * E_DIM + h * D_H + ct * 16 +
          lrow] = o[ct][i] * invl[i];
}

// ---------------------------------------------------------------------------
extern "C" void kernel_launch(void* const* d_in, const int* in_sizes, int n_in,
                              void* d_out, int out_size, void* d_ws,
                              size_t ws_size, hipStream_t stream) {
  (void)in_sizes; (void)n_in; (void)out_size; (void)ws_size;
  const float* values = (const float*)d_in[0];
  const float* keys   = (const float*)d_in[1];
  const float* query  = (const float*)d_in[2];
  const int*   mask   = (const int*)d_in[3];
  const float* Wv = (const float*)d_in[4];
  const float* Wk = (const float*)d_in[5];
  const float* Wq = (const float*)d_in[6];
  const float* Wo = (const float*)d_in[7];
  const float* bo = (const float*)d_in[8];
  float* out = (float*)d_out;

  const size_t bufElems = (size_t)N_B * S_DIM * E_DIM;
  float* qbuf = (float*)d_ws;
  float* kbuf = qbuf + bufElems;
  float* vbuf = kbuf + bufElems;
  float* cbuf = vbuf + bufElems;

  const dim3 gblk(256);
  const dim3 ggrid((N_B * S_DIM) / 128, E_DIM / 128);
  gemm_xwt_kernel<<<ggrid, gblk, 0, stream>>>(query,  Wq, qbuf, nullptr);
  gemm_xwt_kernel<<<ggrid, gblk, 0, stream>>>(keys,   Wk, kbuf, nullptr);
  gemm_xwt_kernel<<<ggrid, gblk, 0, stream>>>(values, Wv, vbuf, nullptr);

  attention_kernel<<<dim3(S_DIM / 64, H_N, N_B), dim3(128), 0, stream>>>(
      qbuf, kbuf, vbuf, mask, cbuf);

  gemm_xwt_kernel<<<ggrid, gblk, 0, stream>>>(cbuf, Wo, out, bo);
}